// RW_NN_32530082300305
// MI455X (gfx1250) — compile-verified
//
#include <hip/hip_runtime.h>
#include <hip/hip_bf16.h>

#define N_NODES 100000
#define N_EDGES 1600000
#define G_GRAPHS 512
#define IN_DIM 128
#define HID 64
#define HGC 16          // hidden graphs
#define KH 8            // size hidden graphs
#define GK 128          // HGC*KH
#define HCOLS 48        // HGC*MAX_STEP
#define NT (N_NODES/16) // 6250 node tiles
#define W1 4            // waves per block in WMMA kernels

typedef __attribute__((ext_vector_type(2))) float v2f;
typedef __attribute__((ext_vector_type(8))) float v8f;

__device__ __forceinline__ v8f wmma4(v2f a, v2f b, v8f c) {
  // D = A(16x4 f32) * B(4x16 f32) + C(16x16 f32)
  return __builtin_amdgcn_wmma_f32_16x16x4_f32(false, a, false, b, (short)0, c,
                                               false, false);
}

// ---------------------------------------------------------------------------
// Kernel 1: x0 = sigmoid(F @ fc_w^T + b)  (WMMA, K=128)
//           zx = x0 @ Z^T                 (WMMA, K=64, A restaged via LDS)
//           step-0 pooled: sum_k zx^2 per (node,hg) -> atomic h_acc[:, 0:16]
//           counts[g] += 1 per node
// One wave handles a 16-node tile. Wave32, EXEC all ones inside uniform guard.
// ---------------------------------------------------------------------------
__global__ __launch_bounds__(W1 * 32) void node_transform(
    const float* __restrict__ feat, const float* __restrict__ fcw,
    const float* __restrict__ fcb, const float* __restrict__ fh,
    const int* __restrict__ gind, float* __restrict__ x0,
    float* __restrict__ zx, float* __restrict__ h_acc,
    float* __restrict__ counts) {
  __shared__ float lx[W1 * 16 * HID];  // 16KB : sigmoid(x) tiles
  __shared__ float lt[W1 * 16 * GK];   // 32KB : zx tiles

  const int wave = threadIdx.x >> 5;
  const int lane = threadIdx.x & 31;
  const int half = lane >> 4;
  const int l16 = lane & 15;
  const int tile = blockIdx.x * W1 + wave;
  const bool act = tile < NT;
  const int nb = tile * 16;
  float* wx = lx + wave * 16 * HID;
  float* wt = lt + wave * 16 * GK;

  const v8f vz = {0.f, 0.f, 0.f, 0.f, 0.f, 0.f, 0.f, 0.f};
  v8f c1[4] = {vz, vz, vz, vz};

  if (act) {
    // GEMM1: (16 x 128) @ (128 x 64) via 32 K-steps x 4 N-tiles of WMMA
    for (int k = 0; k < IN_DIM; k += 4) {
      v2f a = *(const v2f*)(feat + (size_t)(nb + l16) * IN_DIM + k + 2 * half);
#pragma unroll
      for (int j = 0; j < 4; ++j) {
        v2f b = *(const v2f*)(fcw + (size_t)(j * 16 + l16) * IN_DIM + k + 2 * half);
        c1[j] = wmma4(a, b, c1[j]);
      }
    }
    // bias + sigmoid, stage into LDS (C layout: vgpr i -> row i + 8*half)
#pragma unroll
    for (int j = 0; j < 4; ++j) {
      float bj = fcb[j * 16 + l16];
#pragma unroll
      for (int i = 0; i < 8; ++i) {
        float v = c1[j][i] + bj;
        v = 1.0f / (1.0f + __expf(-v));
        wx[(i + 8 * half) * HID + j * 16 + l16] = v;
      }
    }
  }
  __syncthreads();

  v8f c2[8] = {vz, vz, vz, vz, vz, vz, vz, vz};
  if (act) {
    // coalesced x0 store (needed by SpMM step 1)
#pragma unroll
    for (int r = 0; r < 8; ++r) {
      int row = r * 2 + half;
      float4 v = *(const float4*)(wx + row * HID + l16 * 4);
      *(float4*)(x0 + (size_t)(nb + row) * HID + l16 * 4) = v;
    }
    // GEMM2: (16 x 64) @ (64 x 128): A-fragments from LDS, B = features_hidden
    for (int k = 0; k < HID; k += 4) {
      v2f a;
      a.x = wx[l16 * HID + k + 2 * half];
      a.y = wx[l16 * HID + k + 2 * half + 1];
#pragma unroll
      for (int t = 0; t < 8; ++t) {
        v2f b = *(const v2f*)(fh + (size_t)(t * 16 + l16) * HID + k + 2 * half);
        c2[t] = wmma4(a, b, c2[t]);
      }
    }
    // stage zx tile into LDS
#pragma unroll
    for (int t = 0; t < 8; ++t)
#pragma unroll
      for (int i = 0; i < 8; ++i)
        wt[(i + 8 * half) * GK + t * 16 + l16] = c2[t][i];
  }
  __syncthreads();

  if (act) {
    // coalesced zx store
#pragma unroll
    for (int r = 0; r < 16; ++r) {
      float4 v = *(const float4*)(wt + r * GK + lane * 4);
      *(float4*)(zx + (size_t)(nb + r) * GK + lane * 4) = v;
    }
    if (lane < 16) atomicAdd(&counts[gind[nb + lane]], 1.0f);
    // step-0: s[node,hg] = sum_k zx^2 -> pooled (cols 0..15)
#pragma unroll
    for (int it = 0; it < 8; ++it) {
      int pair = it * 32 + lane;
      int node = pair >> 4, hg = pair & 15;
      const float* q = wt + node * GK + hg * 8;
      float s = 0.f;
#pragma unroll
      for (int k = 0; k < 8; ++k) s += q[k] * q[k];
      atomicAdd(&h_acc[gind[nb + node] * HCOLS + hg], s);
    }
  }
}

// ---------------------------------------------------------------------------
// Steps 1/2: t = x_s @ z_s^T (WMMA K=64), then s[n,hg] = sum_k zx*t -> pooled
// ---------------------------------------------------------------------------
__global__ __launch_bounds__(W1 * 32) void step_gemm(
    const float* __restrict__ xs, const float* __restrict__ zs,
    const float* __restrict__ zx, const int* __restrict__ gind,
    float* __restrict__ h_acc, int col0) {
  __shared__ float lt[W1 * 16 * GK];
  const int wave = threadIdx.x >> 5;
  const int lane = threadIdx.x & 31;
  const int half = lane >> 4;
  const int l16 = lane & 15;
  const int tile = blockIdx.x * W1 + wave;
  const bool act = tile < NT;
  const int nb = tile * 16;
  float* wt = lt + wave * 16 * GK;

  const v8f vz = {0.f, 0.f, 0.f, 0.f, 0.f, 0.f, 0.f, 0.f};
  v8f c2[8] = {vz, vz, vz, vz, vz, vz, vz, vz};
  if (act) {
    for (int k = 0; k < HID; k += 4) {
      v2f a = *(const v2f*)(xs + (size_t)(nb + l16) * HID + k + 2 * half);
#pragma unroll
      for (int t = 0; t < 8; ++t) {
        v2f b = *(const v2f*)(zs + (size_t)(t * 16 + l16) * HID + k + 2 * half);
        c2[t] = wmma4(a, b, c2[t]);
      }
    }
#pragma unroll
    for (int t = 0; t < 8; ++t)
#pragma unroll
      for (int i = 0; i < 8; ++i)
        wt[(i + 8 * half) * GK + t * 16 + l16] = c2[t][i];
  }
  __syncthreads();
  if (act) {
#pragma unroll
    for (int it = 0; it < 8; ++it) {
      int pair = it * 32 + lane;
      int node = pair >> 4, hg = pair & 15;
      const float* qt = wt + node * GK + hg * 8;
      const float* qz = zx + (size_t)(nb + node) * GK + hg * 8;
      float s = 0.f;
#pragma unroll
      for (int k = 0; k < 8; ++k) s += qt[k] * qz[k];
      atomicAdd(&h_acc[gind[nb + node] * HCOLS + col0 + hg], s);
    }
  }
}

// ---------------------------------------------------------------------------
// z update: z_out[g,i,h] = sum_j relu-sym-A[g,i,j] * z_in[g,j,h]  (tiny)
// ---------------------------------------------------------------------------
__global__ void update_z(const float* __restrict__ adjh,
                         const float* __restrict__ zin,
                         float* __restrict__ zout) {
  int tid = blockIdx.x * blockDim.x + threadIdx.x;
  if (tid >= HGC * KH * HID) return;
  int g = tid >> 9;
  int i = (tid >> 6) & 7;
  int h = tid & 63;
  float acc = 0.f;
#pragma unroll
  for (int j = 0; j < KH; ++j) {
    if (j == i) continue;
    int a = i < j ? i : j;
    int b = i < j ? j : i;
    int idx = a * 7 - a * (a - 1) / 2 + (b - a - 1);  // triu_indices(8,1) order
    float av = adjh[g * 28 + idx];
    av = av > 0.f ? av : 0.f;
    acc += av * zin[(size_t)g * (KH * HID) + j * HID + h];
  }
  zout[tid] = acc;
}

// ---------------------------------------------------------------------------
// CSR build (avoids 100M float atomics in SpMM)
// ---------------------------------------------------------------------------
__global__ void count_edges(const int* __restrict__ erow, int* __restrict__ cnt) {
  int e = blockIdx.x * blockDim.x + threadIdx.x;
  if (e < N_EDGES) atomicAdd(&cnt[erow[e]], 1);
}

__global__ void scan1(const int* __restrict__ cnt, int* __restrict__ incl,
                      int* __restrict__ bsums) {
  __shared__ int sd[256];
  int t = threadIdx.x;
  int i = blockIdx.x * 256 + t;
  sd[t] = (i < N_NODES) ? cnt[i] : 0;
  __syncthreads();
  for (int off = 1; off < 256; off <<= 1) {
    int a = (t >= off) ? sd[t - off] : 0;
    __syncthreads();
    sd[t] += a;
    __syncthreads();
  }
  if (i < N_NODES) incl[i] = sd[t];
  if (t == 255) bsums[blockIdx.x] = sd[255];
}

__global__ void scan2(const int* __restrict__ bsums, int* __restrict__ boff,
                      int nblk) {
  __shared__ int sd[512];
  int t = threadIdx.x;
  int v = (t < nblk) ? bsums[t] : 0;
  sd[t] = v;
  __syncthreads();
  for (int off = 1; off < 512; off <<= 1) {
    int a = (t >= off) ? sd[t - off] : 0;
    __syncthreads();
    sd[t] += a;
    __syncthreads();
  }
  if (t < nblk) boff[t] = sd[t] - v;  // exclusive
}

__global__ void scan3(const int* __restrict__ cnt, const int* __restrict__ incl,
                      const int* __restrict__ boff, int* __restrict__ rp,
                      int* __restrict__ cursor) {
  int i = blockIdx.x * 256 + threadIdx.x;
  if (i < N_NODES) {
    int ex = incl[i] - cnt[i] + boff[i >> 8];
    rp[i] = ex;
    cursor[i] = ex;
  }
  if (i == 0) rp[N_NODES] = N_EDGES;
}

__global__ void scatter_edges(const int* __restrict__ erow,
                              const int* __restrict__ ecol,
                              const float* __restrict__ ew,
                              int* __restrict__ cursor, int* __restrict__ ccol,
                              float* __restrict__ cw) {
  int e = blockIdx.x * blockDim.x + threadIdx.x;
  if (e >= N_EDGES) return;
  int r = erow[e];
  int p = atomicAdd(&cursor[r], 1);
  ccol[p] = ecol[e];
  cw[p] = ew[e];
}

// wave-per-row gather SpMM: x_out[row,:] = sum_e w_e * x_in[col_e,:]
// x_in (25.6MB) is L2-resident on MI455X (192MB L2).
__global__ __launch_bounds__(256) void spmm(const int* __restrict__ rp,
                                            const int* __restrict__ ccol,
                                            const float* __restrict__ cw,
                                            const float* __restrict__ xin,
                                            float* __restrict__ xout) {
  int wave = threadIdx.x >> 5, lane = threadIdx.x & 31;
  int row = blockIdx.x * 8 + wave;
  if (row >= N_NODES) return;
  int s = rp[row], e = rp[row + 1];
  float2 acc = make_float2(0.f, 0.f);
  for (int i = s; i < e; ++i) {
    int c = ccol[i];
    float w = cw[i];
    float2 v = *(const float2*)(xin + (size_t)c * HID + lane * 2);
    acc.x += w * v.x;
    acc.y += w * v.y;
  }
  *(float2*)(xout + (size_t)row * HID + lane * 2) = acc;
}

// ---------------------------------------------------------------------------
// Head: norm, BN (train stats over G), fc1+relu, fc2, log_softmax. 1 block.
// ---------------------------------------------------------------------------
__global__ __launch_bounds__(512) void final_head(
    const float* __restrict__ h_acc, const float* __restrict__ counts,
    const float* __restrict__ gam, const float* __restrict__ bet,
    const float* __restrict__ f1w, const float* __restrict__ f1b,
    const float* __restrict__ f2w, const float* __restrict__ f2b,
    float* __restrict__ out) {
  __shared__ float ssum[HCOLS], ssq[HCOLS], smean[HCOLS], sinv[HCOLS];
  int g = threadIdx.x;
  if (g < HCOLS) { ssum[g] = 0.f; ssq[g] = 0.f; }
  __syncthreads();
  float h[HCOLS];
  float nrm = fmaxf(counts[g], 1.0f);
#pragma unroll
  for (int c = 0; c < HCOLS; ++c) {
    h[c] = h_acc[g * HCOLS + c] / nrm;
    atomicAdd(&ssum[c], h[c]);
    atomicAdd(&ssq[c], h[c] * h[c]);
  }
  __syncthreads();
  if (g < HCOLS) {
    float m = ssum[g] / (float)G_GRAPHS;
    float v = ssq[g] / (float)G_GRAPHS - m * m;
    smean[g] = m;
    sinv[g] = 1.0f / sqrtf(v + 1e-5f);
  }
  __syncthreads();
  float hn[HCOLS];
#pragma unroll
  for (int c = 0; c < HCOLS; ++c)
    hn[c] = gam[c] * (h[c] - smean[c]) * sinv[c] + bet[c];
  float o[10];
#pragma unroll
  for (int k = 0; k < 10; ++k) o[k] = f2b[k];
  for (int j = 0; j < 128; ++j) {
    float p = f1b[j];
#pragma unroll
    for (int c = 0; c < HCOLS; ++c) p += f1w[j * HCOLS + c] * hn[c];
    p = fmaxf(p, 0.f);
#pragma unroll
    for (int k = 0; k < 10; ++k) o[k] += f2w[k * 128 + j] * p;
  }
  float m = o[0];
#pragma unroll
  for (int k = 1; k < 10; ++k) m = fmaxf(m, o[k]);
  float s = 0.f;
#pragma unroll
  for (int k = 0; k < 10; ++k) s += expf(o[k] - m);
  float lse = m + logf(s);
#pragma unroll
  for (int k = 0; k < 10; ++k) out[g * 10 + k] = o[k] - lse;
}

extern "C" void kernel_launch(void* const* d_in, const int* in_sizes, int n_in,
                              void* d_out, int out_size, void* d_ws,
                              size_t ws_size, hipStream_t stream) {
  const float* feat = (const float*)d_in[0];
  const float* ew = (const float*)d_in[1];
  const float* adjh = (const float*)d_in[2];
  const float* fh = (const float*)d_in[3];
  const float* fcw = (const float*)d_in[4];
  const float* fcb = (const float*)d_in[5];
  const float* gam = (const float*)d_in[6];
  const float* bet = (const float*)d_in[7];
  const float* f1w = (const float*)d_in[8];
  const float* f1b = (const float*)d_in[9];
  const float* f2w = (const float*)d_in[10];
  const float* f2b = (const float*)d_in[11];
  const int* gind = (const int*)d_in[12];
  const int* erow = (const int*)d_in[13];
  const int* ecol = (const int*)d_in[14];
  float* out = (float*)d_out;

  char* ws = (char*)d_ws;
  size_t off = 0;
  auto take = [&](size_t bytes) -> void* {
    void* p = ws + off;
    off = (off + bytes + 255) & ~(size_t)255;
    return p;
  };
  float* zx = (float*)take((size_t)N_NODES * GK * 4);
  float* x0 = (float*)take((size_t)N_NODES * HID * 4);
  float* x1 = (float*)take((size_t)N_NODES * HID * 4);
  int* row_cnt = (int*)take((size_t)N_NODES * 4);
  int* tincl = (int*)take((size_t)N_NODES * 4);
  int* rp = (int*)take((size_t)(N_NODES + 1) * 4);
  int* cursor = (int*)take((size_t)N_NODES * 4);
  int* bsums = (int*)take(512 * 4);
  int* boff = (int*)take(512 * 4);
  int* ccol = (int*)take((size_t)N_EDGES * 4);
  float* cw = (float*)take((size_t)N_EDGES * 4);
  float* z1 = (float*)take((size_t)HGC * KH * HID * 4);
  float* z2 = (float*)take((size_t)HGC * KH * HID * 4);
  float* hacc = (float*)take((size_t)G_GRAPHS * HCOLS * 4);
  float* cnts = (float*)take((size_t)G_GRAPHS * 4);

  hipMemsetAsync(row_cnt, 0, (size_t)N_NODES * 4, stream);
  hipMemsetAsync(hacc, 0, (size_t)G_GRAPHS * HCOLS * 4, stream);
  hipMemsetAsync(cnts, 0, (size_t)G_GRAPHS * 4, stream);

  const int nblk1 = (N_NODES + 255) / 256;           // 391
  const int gblk = (NT + W1 - 1) / W1;               // 1563

  update_z<<<(HGC * KH * HID + 255) / 256, 256, 0, stream>>>(adjh, fh, z1);
  update_z<<<(HGC * KH * HID + 255) / 256, 256, 0, stream>>>(adjh, z1, z2);

  node_transform<<<gblk, W1 * 32, 0, stream>>>(feat, fcw, fcb, fh, gind, x0, zx,
                                               hacc, cnts);

  count_edges<<<(N_EDGES + 255) / 256, 256, 0, stream>>>(erow, row_cnt);
  scan1<<<nblk1, 256, 0, stream>>>(row_cnt, tincl, bsums);
  scan2<<<1, 512, 0, stream>>>(bsums, boff, nblk1);
  scan3<<<nblk1, 256, 0, stream>>>(row_cnt, tincl, boff, rp, cursor);
  scatter_edges<<<(N_EDGES + 255) / 256, 256, 0, stream>>>(erow, ecol, ew,
                                                           cursor, ccol, cw);

  spmm<<<N_NODES / 8, 256, 0, stream>>>(rp, ccol, cw, x0, x1);
  step_gemm<<<gblk, W1 * 32, 0, stream>>>(x1, z1, zx, gind, hacc, 16);
  spmm<<<N_NODES / 8, 256, 0, stream>>>(rp, ccol, cw, x1, x0);  // x0 = x2
  step_gemm<<<gblk, W1 * 32, 0, stream>>>(x0, z2, zx, gind, hacc, 32);

  final_head<<<1, G_GRAPHS, 0, stream>>>(hacc, cnts, gam, bet, f1w, f1b, f2w,
                                         f2b, out);
}